// DynamicEmbedder_627065225425
// MI455X (gfx1250) — compile-verified
//
#include <hip/hip_runtime.h>

typedef float v2f __attribute__((ext_vector_type(2)));
typedef float v8f __attribute__((ext_vector_type(8)));

#define GNX   512
#define GNY   512
#define GNSEG (GNX * GNY)
#define GC    64
#define BN_EPS 1e-3f

// grid constants
#define PMINX (-51.2f)
#define PMINY (-51.2f)
#define PMINZ (-3.0f)
#define PMAXX (51.2f)
#define PMAXY (51.2f)
#define PMAXZ (3.0f)
#define VOXX  (0.2f)
#define VOXY  (0.2f)
#define VOXZ  (6.0f)

// ---------------------------------------------------------------------------
// Pass 1: per-pillar count + xyz sums (atomic scatter; counts are exact ints)
// ---------------------------------------------------------------------------
__global__ void pillar_stats_kernel(const float* __restrict__ pts,
                                    float* __restrict__ cnt,
                                    float* __restrict__ sx,
                                    float* __restrict__ sy,
                                    float* __restrict__ sz,
                                    int N, int B) {
    int i = blockIdx.x * blockDim.x + threadIdx.x;
    int total = B * N;
    if (i >= total) return;
    int b = i / N;
    float px = pts[3 * i + 0];
    float py = pts[3 * i + 1];
    float pz = pts[3 * i + 2];
    bool inr = (px >= PMINX) && (px < PMAXX) &&
               (py >= PMINY) && (py < PMAXY) &&
               (pz >= PMINZ) && (pz < PMAXZ);
    if (!inr) return;
    int cx = (int)floorf((px - PMINX) / VOXX);
    int cy = (int)floorf((py - PMINY) / VOXY);
    int s = b * GNSEG + cy * GNX + cx;
    atomicAdd(&cnt[s], 1.0f);
    atomicAdd(&sx[s], px);
    atomicAdd(&sy[s], py);
    atomicAdd(&sz[s], pz);
}

// ---------------------------------------------------------------------------
// Pass 2: features -> Linear(9->64) via V_WMMA_F32_16X16X4_F32 -> BN -> ReLU
//         -> masked atomic scatter-sum into dense canvas [C, NY, NX]
// One wave (32 lanes) owns a tile of 16 points; grid-stride over tiles.
// A (16x4 f32):  VGPR0 = {K0 (lanes 0-15), K2 (lanes 16-31)}, VGPR1 = {K1, K3}
// B (4x16 f32):  VGPR0 = {row0 (lo), row2 (hi)}, VGPR1 = {row1 (lo), row3 (hi)}
// D (16x16 f32): VGPR r -> M = r (lo lanes) / 8+r (hi lanes), N = lane & 15
// ---------------------------------------------------------------------------
__global__ void pfn_wmma_kernel(const float* __restrict__ pts,
                                const float* __restrict__ W,      // [9, 64]
                                const float* __restrict__ gamma,
                                const float* __restrict__ beta,
                                const float* __restrict__ rmean,
                                const float* __restrict__ rvar,
                                const float* __restrict__ cnt,
                                const float* __restrict__ sx,
                                const float* __restrict__ sy,
                                const float* __restrict__ sz,
                                float* __restrict__ out,          // [B, C, NSEG]
                                int N, int tilesPerB, int numTiles) {
    const int lane = threadIdx.x & 31;
    const bool hi  = lane >= 16;
    const int ml   = lane & 15;
    const int waveId = (blockIdx.x * blockDim.x + threadIdx.x) >> 5;
    const int nWaves = (gridDim.x * blockDim.x) >> 5;

    // --- load W fragments (K padded 9->12, zeros) + BN affine, once per wave ---
    v2f bf[4][3];
    float scale[4], bias[4];
#pragma unroll
    for (int nt = 0; nt < 4; ++nt) {
        const int c = nt * 16 + ml;
        int r0 = hi ? 2 : 0, r1 = hi ? 3 : 1;
        bf[nt][0].x = W[r0 * GC + c];
        bf[nt][0].y = W[r1 * GC + c];
        r0 = hi ? 6 : 4; r1 = hi ? 7 : 5;
        bf[nt][1].x = W[r0 * GC + c];
        bf[nt][1].y = W[r1 * GC + c];
        bf[nt][2].x = hi ? 0.0f : W[8 * GC + c];   // row 8 valid, rows 9..11 pad
        bf[nt][2].y = 0.0f;
        const float sc = gamma[c] * rsqrtf(rvar[c] + BN_EPS);
        scale[nt] = sc;
        bias[nt]  = beta[c] - rmean[c] * sc;
    }

    for (int t = waveId; t < numTiles; t += nWaves) {       // wave-uniform loop
        const int b    = t / tilesPerB;
        const int base = (t - b * tilesPerB) * 16;
        const int pIdx = base + ml;                          // this lane's point (M = ml)
        const bool valid = pIdx < N;
        const int g = b * N + (valid ? pIdx : 0);

        const float px = pts[3 * g + 0];
        const float py = pts[3 * g + 1];
        const float pz = pts[3 * g + 2];

        const float cxf = floorf((px - PMINX) / VOXX);
        const float cyf = floorf((py - PMINY) / VOXY);
        const float czf = floorf((pz - PMINZ) / VOXZ);
        const bool inr = valid &&
                         (px >= PMINX) && (px < PMAXX) &&
                         (py >= PMINY) && (py < PMAXY) &&
                         (pz >= PMINZ) && (pz < PMAXZ);
        const int  pid = inr ? ((int)cyf * GNX + (int)cxf) : 0;
        const float wm = inr ? 1.0f : 0.0f;

        const int sidx = b * GNSEG + pid;
        const float inv = 1.0f / fmaxf(cnt[sidx], 1.0f);
        const float mx = sx[sidx] * inv;
        const float my = sy[sidx] * inv;
        const float mz = sz[sidx] * inv;

        // 9 features: xyz, cluster offset, voxel-center offset
        const float f0 = px, f1 = py, f2 = pz;
        const float f3 = px - mx, f4 = py - my, f5 = pz - mz;
        const float f6 = px - ((cxf + 0.5f) * VOXX + PMINX);
        const float f7 = py - ((cyf + 0.5f) * VOXY + PMINY);
        const float f8 = pz - ((czf + 0.5f) * VOXZ + PMINZ);

        // A fragments (3 K-chunks of 4)
        v2f a0, a1, a2;
        a0.x = hi ? f2 : f0;  a0.y = hi ? f3 : f1;
        a1.x = hi ? f6 : f4;  a1.y = hi ? f7 : f5;
        a2.x = hi ? 0.0f : f8; a2.y = 0.0f;

        // fetch pid/mask of the 8 rows this half-wave owns in D
        const int mbase = hi ? 8 : 0;
        int   pidArr[8];
        float wArr[8];
#pragma unroll
        for (int r = 0; r < 8; ++r) {
            pidArr[r] = __shfl(pid, mbase + r, 32);
            wArr[r]   = __shfl(wm,  mbase + r, 32);
        }

        float* const outB = out + (size_t)b * (GC * GNSEG);
#pragma unroll
        for (int nt = 0; nt < 4; ++nt) {
            v8f acc = {};
            acc = __builtin_amdgcn_wmma_f32_16x16x4_f32(false, a0, false, bf[nt][0],
                                                        (short)0, acc, false, false);
            acc = __builtin_amdgcn_wmma_f32_16x16x4_f32(false, a1, false, bf[nt][1],
                                                        (short)0, acc, false, false);
            acc = __builtin_amdgcn_wmma_f32_16x16x4_f32(false, a2, false, bf[nt][2],
                                                        (short)0, acc, false, false);
            const int c = nt * 16 + ml;
            float* const outC = outB + (size_t)c * GNSEG;
#pragma unroll
            for (int r = 0; r < 8; ++r) {
                const float v = fmaxf(acc[r] * scale[nt] + bias[nt], 0.0f);
                if (wArr[r] > 0.0f)
                    atomicAdd(outC + pidArr[r], v);
            }
        }
    }
}

// ---------------------------------------------------------------------------
// Pass 3: per-pillar average (in place), float4 vectorized, cnt hits L2
// ---------------------------------------------------------------------------
__global__ void normalize_kernel(float* __restrict__ out,
                                 const float* __restrict__ cnt,
                                 int total4) {   // total elements / 4
    int i = blockIdx.x * blockDim.x + threadIdx.x;
    const int stride = gridDim.x * blockDim.x;
    for (; i < total4; i += stride) {
        const int e  = i << 2;
        const int s  = e & (GNSEG - 1);          // NSEG is a power of two
        const int bc = e >> 18;                  // e / NSEG
        const int b  = bc >> 6;                  // bc / C
        const float* cp = cnt + b * GNSEG + s;
        float4 v = reinterpret_cast<float4*>(out)[i];
        v.x /= fmaxf(cp[0], 1.0f);
        v.y /= fmaxf(cp[1], 1.0f);
        v.z /= fmaxf(cp[2], 1.0f);
        v.w /= fmaxf(cp[3], 1.0f);
        reinterpret_cast<float4*>(out)[i] = v;
    }
}

extern "C" void kernel_launch(void* const* d_in, const int* in_sizes, int n_in,
                              void* d_out, int out_size, void* d_ws, size_t ws_size,
                              hipStream_t stream) {
    const float* points = (const float*)d_in[0];
    const float* W      = (const float*)d_in[1];
    const float* gamma  = (const float*)d_in[2];
    const float* beta   = (const float*)d_in[3];
    const float* rmean  = (const float*)d_in[4];
    const float* rvar   = (const float*)d_in[5];
    float* out = (float*)d_out;

    const int B = 2;
    const int N = in_sizes[0] / (B * 3);         // 100000

    float* cnt = (float*)d_ws;                   // [B*NSEG]
    float* sx  = cnt + (size_t)B * GNSEG;
    float* sy  = sx  + (size_t)B * GNSEG;
    float* sz  = sy  + (size_t)B * GNSEG;

    hipMemsetAsync(d_out, 0, (size_t)out_size * sizeof(float), stream);
    hipMemsetAsync(d_ws, 0, (size_t)4 * B * GNSEG * sizeof(float), stream);

    // pass 1: pillar stats
    {
        const int total = B * N;
        const int thr = 256;
        pillar_stats_kernel<<<(total + thr - 1) / thr, thr, 0, stream>>>(
            points, cnt, sx, sy, sz, N, B);
    }

    // pass 2: WMMA PFN + scatter
    {
        const int tilesPerB = (N + 15) / 16;
        const int numTiles  = B * tilesPerB;     // 12500
        const int thr = 256;                     // 8 waves / block
        const int blk = 200;                     // 1600 waves, ~8 tiles each
        pfn_wmma_kernel<<<blk, thr, 0, stream>>>(
            points, W, gamma, beta, rmean, rvar,
            cnt, sx, sy, sz, out, N, tilesPerB, numTiles);
    }

    // pass 3: average
    {
        const int total4 = (B * GC * GNSEG) >> 2;
        const int thr = 256;
        const int blk = 2048;
        normalize_kernel<<<blk, thr, 0, stream>>>(out, cnt, total4);
    }
}